// GraphVAE_919123001659
// MI455X (gfx1250) — compile-verified
//
#include <hip/hip_runtime.h>
#include <cstdint>

typedef float v2f __attribute__((ext_vector_type(2)));
typedef float v8f __attribute__((ext_vector_type(8)));

static constexpr int kN     = 20000;
static constexpr int kE     = 120000;   // positive edges
static constexpr int kE2    = 240000;   // undirected message edges
static constexpr int kENEG  = 600000;   // negative edges
static constexpr int kETOT  = 720000;   // kE + kENEG
static constexpr int kEGRP  = kETOT / 32;  // 22500 edge groups of 32

__device__ __forceinline__ void atomAddF(float* p, float v) {
  unsafeAtomicAdd(p, v);
}

// ---------------------------------------------------------------------------
// Pack B[K,N] (row-major) into WMMA fragment order:
//   frag f = k0*(N/16) + t   (k0 = K-step, t = N-tile), 64 floats per frag,
//   lane holds { B[4k0+2hi, t*16+lo], B[4k0+2hi+1, t*16+lo] } at BP + f*64 + 2*lane
// One b64 load then yields a whole B fragment.
// ---------------------------------------------------------------------------
__global__ __launch_bounds__(256)
void pack_b_frags(const float* __restrict__ B, float* __restrict__ BP,
                  int K, int Nc) {
  int idx = blockIdx.x * blockDim.x + threadIdx.x;
  int total = (K >> 2) * (Nc >> 4) * 32;
  if (idx >= total) return;
  int lane = idx & 31;
  int f = idx >> 5;
  int nt = Nc >> 4;
  int t = f % nt;
  int k0 = f / nt;
  int lo = lane & 15, hi = lane >> 4;
  int row = 4 * k0 + 2 * hi;
  int col = t * 16 + lo;
  v2f v;
  v.x = B[(size_t)row * Nc + col];
  v.y = B[(size_t)(row + 1) * Nc + col];
  *(v2f*)(BP + (size_t)f * 64 + 2 * lane) = v;
}

// ---------------------------------------------------------------------------
// fp32 GEMM C[M,N] = A[M,K] @ B[K,N] via V_WMMA_F32_16X16X4_F32.
// B pre-packed in fragment order. One wave computes a 32x16 tile of C
// (2 M-tiles share each B fragment). Requires M%32==0, N%16==0, K%4==0.
// ---------------------------------------------------------------------------
__global__ __launch_bounds__(128)
void gemm_wmma_f32(const float* __restrict__ A, const float* __restrict__ BP,
                   float* __restrict__ C, int M, int Nc, int K) {
  const int lane = threadIdx.x & 31;
  const int wave = blockIdx.x * (blockDim.x >> 5) + (threadIdx.x >> 5);
  const int nt = Nc >> 4;
  const int tiles = (M >> 5) * nt;
  if (wave >= tiles) return;                 // wave-uniform: EXEC stays all-1s
  const int mPair = wave / nt;
  const int nTile = wave - mPair * nt;
  const int lo = lane & 15, hi = lane >> 4;

  const float* a0 = A + (size_t)(mPair * 32 + lo) * K + 2 * hi;
  const float* a1 = a0 + (size_t)16 * K;
  const float* bp = BP + (size_t)nTile * 64 + 2 * lane;
  const size_t bStep = (size_t)nt * 64;

  v8f c0 = {}, c1 = {};
  const int ksteps = K >> 2;
  for (int k0 = 0; k0 < ksteps; ++k0) {
    v2f b  = *(const v2f*)(bp + (size_t)k0 * bStep);
    v2f A0 = *(const v2f*)(a0 + 4 * k0);
    v2f A1 = *(const v2f*)(a1 + 4 * k0);
    c0 = __builtin_amdgcn_wmma_f32_16x16x4_f32(false, A0, false, b, (short)0, c0, false, false);
    c1 = __builtin_amdgcn_wmma_f32_16x16x4_f32(false, A1, false, b, (short)0, c1, false, false);
  }
  // C layout: VGPR r -> C[m0 + r + 8*hi, n0 + lo]
  float* cp0 = C + (size_t)(mPair * 32 + 8 * hi) * Nc + nTile * 16 + lo;
  float* cp1 = cp0 + (size_t)16 * Nc;
#pragma unroll
  for (int r = 0; r < 8; ++r) cp0[(size_t)r * Nc] = c0[r];
#pragma unroll
  for (int r = 0; r < 8; ++r) cp1[(size_t)r * Nc] = c1[r];
}

// ---------------------------------------------------------------------------
// Encoder scatter-add: agg[dst] += m[src], F=256 (4 floats per thread)
// ---------------------------------------------------------------------------
__global__ __launch_bounds__(256)
void scatter_add_256(const float* __restrict__ m, float* __restrict__ agg,
                     const int* __restrict__ src, const int* __restrict__ dst) {
  int idx = blockIdx.x * blockDim.x + threadIdx.x;
  if (idx >= kE2 * 64) return;
  int e = idx >> 6;
  int c = (idx & 63) << 2;
  int s = src[e], d = dst[e];
  const float4 v = *(const float4*)(m + (size_t)s * 256 + c);
  float* p = agg + (size_t)d * 256 + c;
  atomAddF(p + 0, v.x); atomAddF(p + 1, v.y);
  atomAddF(p + 2, v.z); atomAddF(p + 3, v.w);
}

// LayerNorm(row + b1)*g1 + bt1, then ReLU; one wave per 256-wide row, in place.
__global__ __launch_bounds__(256)
void ln_relu_rows(float* __restrict__ buf, const float* __restrict__ b1,
                  const float* __restrict__ g1, const float* __restrict__ bt1) {
  int lane = threadIdx.x & 31;
  int row = blockIdx.x * 8 + (threadIdx.x >> 5);
  if (row >= kN) return;
  float* p = buf + (size_t)row * 256;
  float v[8];
  float sum = 0.f;
#pragma unroll
  for (int i = 0; i < 8; ++i) {
    int c = lane * 8 + i;
    v[i] = p[c] + b1[c];
    sum += v[i];
  }
#pragma unroll
  for (int off = 1; off < 32; off <<= 1) sum += __shfl_xor(sum, off);
  float mean = sum * (1.0f / 256.0f);
  float vs = 0.f;
#pragma unroll
  for (int i = 0; i < 8; ++i) { float d = v[i] - mean; vs += d * d; }
#pragma unroll
  for (int off = 1; off < 32; off <<= 1) vs += __shfl_xor(vs, off);
  float rstd = rsqrtf(vs * (1.0f / 256.0f) + 1e-5f);
#pragma unroll
  for (int i = 0; i < 8; ++i) {
    int c = lane * 8 + i;
    p[c] = fmaxf((v[i] - mean) * rstd * g1[c] + bt1[c], 0.f);
  }
}

// mu/logvar scatter-add together, F=64
__global__ __launch_bounds__(256)
void scatter_add_64x2(const float* __restrict__ mm, const float* __restrict__ lm,
                      float* __restrict__ muA, float* __restrict__ lvA,
                      const int* __restrict__ src, const int* __restrict__ dst) {
  int idx = blockIdx.x * blockDim.x + threadIdx.x;
  if (idx >= kE2 * 16) return;
  int e = idx >> 4;
  int c = (idx & 15) << 2;
  int s = src[e], d = dst[e];
  float4 a = *(const float4*)(mm + (size_t)s * 64 + c);
  float4 b = *(const float4*)(lm + (size_t)s * 64 + c);
  float* pa = muA + (size_t)d * 64 + c;
  float* pb = lvA + (size_t)d * 64 + c;
  atomAddF(pa + 0, a.x); atomAddF(pa + 1, a.y);
  atomAddF(pa + 2, a.z); atomAddF(pa + 3, a.w);
  atomAddF(pb + 0, b.x); atomAddF(pb + 1, b.y);
  atomAddF(pb + 2, b.z); atomAddF(pb + 3, b.w);
}

// z = mu + eps*exp(0.5*lv); accumulate KL sum of (1 + lv - mu^2 - exp(lv))
__global__ __launch_bounds__(256)
void z_and_kl(const float* __restrict__ muA, const float* __restrict__ lvA,
              const float* __restrict__ eps, const float* __restrict__ bmu,
              const float* __restrict__ blv, float* __restrict__ z,
              float* __restrict__ acc) {
  int idx = blockIdx.x * blockDim.x + threadIdx.x;
  float t = 0.f;
  if (idx < kN * 64) {
    int c = idx & 63;
    float mu = muA[idx] + bmu[c];
    float lv = lvA[idx] + blv[c];
    z[idx] = fmaf(eps[idx], expf(0.5f * lv), mu);
    t = 1.0f + lv - mu * mu - expf(lv);
  }
#pragma unroll
  for (int off = 1; off < 32; off <<= 1) t += __shfl_xor(t, off);
  if ((threadIdx.x & 31) == 0) atomAddF(&acc[0], t);
}

__global__ __launch_bounds__(256)
void deg_count(const int* __restrict__ pd, float* __restrict__ dis) {
  int e = blockIdx.x * blockDim.x + threadIdx.x;
  if (e < kE) atomAddF(&dis[pd[e]], 1.0f);
}

__global__ __launch_bounds__(256)
void deg_finish(float* __restrict__ dis) {
  int i = blockIdx.x * blockDim.x + threadIdx.x;
  if (i < kN) dis[i] = rsqrtf(dis[i] + 1.0f);   // +1 self loop; always >= 1
}

// out[pd] += md[ps] * dis[ps]*dis[pd]   (F=64, 4 floats/thread)
__global__ __launch_bounds__(256)
void scatter_norm64(const float* __restrict__ md, float* __restrict__ out,
                    const int* __restrict__ ps, const int* __restrict__ pd,
                    const float* __restrict__ dis) {
  int idx = blockIdx.x * blockDim.x + threadIdx.x;
  if (idx >= kE * 16) return;
  int e = idx >> 4;
  int c = (idx & 15) << 2;
  int s = ps[e], d = pd[e];
  float w = dis[s] * dis[d];
  float4 v = *(const float4*)(md + (size_t)s * 64 + c);
  float* p = out + (size_t)d * 64 + c;
  atomAddF(p + 0, v.x * w); atomAddF(p + 1, v.y * w);
  atomAddF(p + 2, v.z * w); atomAddF(p + 3, v.w * w);
}

// out = relu(out + md*dis^2 + bias)  (self loop + bias + ReLU)
__global__ __launch_bounds__(256)
void self_bias_relu(float* __restrict__ out, const float* __restrict__ md,
                    const float* __restrict__ dis, const float* __restrict__ bias) {
  int idx = blockIdx.x * blockDim.x + threadIdx.x;
  if (idx >= kN * 64) return;
  int c = idx & 63, i = idx >> 6;
  float w = dis[i];
  out[idx] = fmaxf(out[idx] + md[idx] * w * w + bias[c], 0.f);
}

// ---------------------------------------------------------------------------
// Edge MLP: 32 edges per wave (2 M-tiles sharing packed B fragments).
// phi=[hu|hv||hu-hv||hu*hv] (32x256) staged in LDS, act = relu(phi@Wa + ba)
// via fp32 WMMA, logit = act.Wb + bb, weighted BCE accumulated into acc[1].
// ---------------------------------------------------------------------------
__global__ __launch_bounds__(32)
void edge_mlp_wmma(const float* __restrict__ h, const int* __restrict__ pos,
                   const int* __restrict__ neg, const float* __restrict__ WaP,
                   const float* __restrict__ ba, const float* __restrict__ Wb,
                   const float* __restrict__ bbp, const float* __restrict__ taup,
                   float* __restrict__ acc) {
  constexpr int ROWP = 260;                 // padded row stride (bank spread)
  __shared__ float sPhi[32][ROWP];          // 33,280 B, one wave per block
  const int lane = threadIdx.x & 31;
  const int g = blockIdx.x;                 // edge group of 32
  const int lo = lane & 15, hi = lane >> 4;

  // stage phi rows: 2 floats of hu/hv per lane per edge
#pragma unroll 1
  for (int e = 0; e < 32; ++e) {
    int ge = g * 32 + e;
    int u, v;
    if (ge < kE) { u = pos[ge]; v = pos[kE + ge]; }
    else { int t = ge - kE; u = neg[t]; v = neg[kENEG + t]; }
    v2f hu = *(const v2f*)(h + (size_t)u * 64 + 2 * lane);
    v2f hv = *(const v2f*)(h + (size_t)v * 64 + 2 * lane);
    float* row = &sPhi[e][0];
    *(v2f*)(row + 2 * lane) = hu;
    *(v2f*)(row + 64 + 2 * lane) = hv;
    v2f dd; dd.x = fabsf(hu.x - hv.x); dd.y = fabsf(hu.y - hv.y);
    *(v2f*)(row + 128 + 2 * lane) = dd;
    v2f pp; pp.x = hu.x * hv.x; pp.y = hu.y * hv.y;
    *(v2f*)(row + 192 + 2 * lane) = pp;
  }
  __syncthreads();

  v8f a00 = {}, a01 = {}, a02 = {}, a03 = {};   // M-tile 0, N-tiles 0..3
  v8f a10 = {}, a11 = {}, a12 = {}, a13 = {};   // M-tile 1
  const float* aR0 = &sPhi[0][0] + lo * ROWP + 2 * hi;       // rows 0..15
  const float* aR1 = aR0 + 16 * ROWP;                        // rows 16..31
  const float* wp = WaP + 2 * lane;   // frag (k0*4+t): stride 64 floats
  for (int k0 = 0; k0 < 64; ++k0) {
    const float* wk = wp + (size_t)(k0 * 4) * 64;
    v2f b0 = *(const v2f*)(wk);
    v2f b1 = *(const v2f*)(wk + 64);
    v2f b2 = *(const v2f*)(wk + 128);
    v2f b3 = *(const v2f*)(wk + 192);
    v2f A0 = *(const v2f*)(aR0 + 4 * k0);
    v2f A1 = *(const v2f*)(aR1 + 4 * k0);
    a00 = __builtin_amdgcn_wmma_f32_16x16x4_f32(false, A0, false, b0, (short)0, a00, false, false);
    a10 = __builtin_amdgcn_wmma_f32_16x16x4_f32(false, A1, false, b0, (short)0, a10, false, false);
    a01 = __builtin_amdgcn_wmma_f32_16x16x4_f32(false, A0, false, b1, (short)0, a01, false, false);
    a11 = __builtin_amdgcn_wmma_f32_16x16x4_f32(false, A1, false, b1, (short)0, a11, false, false);
    a02 = __builtin_amdgcn_wmma_f32_16x16x4_f32(false, A0, false, b2, (short)0, a02, false, false);
    a12 = __builtin_amdgcn_wmma_f32_16x16x4_f32(false, A1, false, b2, (short)0, a12, false, false);
    a03 = __builtin_amdgcn_wmma_f32_16x16x4_f32(false, A0, false, b3, (short)0, a03, false, false);
    a13 = __builtin_amdgcn_wmma_f32_16x16x4_f32(false, A1, false, b3, (short)0, a13, false, false);
  }

  // bias + ReLU + dot with Wb, in C layout (row m = mt*16 + r + 8*hi, col t*16+lo)
  float part[2][8];
#pragma unroll
  for (int r = 0; r < 8; ++r) { part[0][r] = 0.f; part[1][r] = 0.f; }
  v8f accs[2][4] = {{a00, a01, a02, a03}, {a10, a11, a12, a13}};
#pragma unroll
  for (int t = 0; t < 4; ++t) {
    float bav = ba[t * 16 + lo];
    float wbv = Wb[t * 16 + lo];
#pragma unroll
    for (int mt = 0; mt < 2; ++mt) {
#pragma unroll
      for (int r = 0; r < 8; ++r) {
        float x = fmaxf(accs[mt][t][r] + bav, 0.f);
        part[mt][r] += x * wbv;
      }
    }
  }
  // reduce across the 16 lanes of each half (cols 0..63)
#pragma unroll
  for (int mt = 0; mt < 2; ++mt) {
#pragma unroll
    for (int r = 0; r < 8; ++r) {
      part[mt][r] += __shfl_xor(part[mt][r], 1);
      part[mt][r] += __shfl_xor(part[mt][r], 2);
      part[mt][r] += __shfl_xor(part[mt][r], 4);
      part[mt][r] += __shfl_xor(part[mt][r], 8);
    }
  }
  if (lo == 0) {                 // lane 0: rows r+0; lane 16: rows r+8 (per mt)
    float bbv = bbp[0];
    float inv = 1.0f / fmaxf(taup[0], 1e-4f);
    bool isPos = (g * 32) < kE;  // kE % 32 == 0: groups never straddle
    float s = 0.f;
#pragma unroll
    for (int mt = 0; mt < 2; ++mt) {
#pragma unroll
      for (int r = 0; r < 8; ++r) {
        float logit = (part[mt][r] + bbv) * inv;
        float xx = isPos ? logit : -logit;
        float ls = fminf(xx, 0.f) - log1pf(expf(-fabsf(xx)));   // log_sigmoid
        s += isPos ? 5.0f * ls : ls;   // pos_weight = NEG = 5
      }
    }
    atomAddF(&acc[1], s);
  }
}

__global__ void finalize_loss(const float* __restrict__ acc, float* __restrict__ out) {
  if (threadIdx.x == 0 && blockIdx.x == 0) {
    float kl = -0.5f * acc[0] / (float)(kN * 64);
    float recon = -acc[1] / (float)kETOT;
    out[0] = recon + kl;   // loss (BETA = 1)
    out[1] = recon;
    out[2] = kl;
  }
}

// ---------------------------------------------------------------------------
extern "C" void kernel_launch(void* const* d_in, const int* in_sizes, int n_in,
                              void* d_out, int out_size, void* d_ws, size_t ws_size,
                              hipStream_t stream) {
  (void)in_sizes; (void)n_in; (void)out_size; (void)ws_size;
  const float* x   = (const float*)d_in[0];
  const float* eps = (const float*)d_in[1];
  const int*   ei  = (const int*)d_in[2];
  const int*   pos = (const int*)d_in[3];
  const int*   neg = (const int*)d_in[4];
  const float* W1  = (const float*)d_in[5];
  const float* b1  = (const float*)d_in[6];
  const float* g1  = (const float*)d_in[7];
  const float* bt1 = (const float*)d_in[8];
  const float* Wmu = (const float*)d_in[9];
  const float* bmu = (const float*)d_in[10];
  const float* Wlv = (const float*)d_in[11];
  const float* blv = (const float*)d_in[12];
  const float* Wd1 = (const float*)d_in[13];
  const float* bd1 = (const float*)d_in[14];
  const float* Wd2 = (const float*)d_in[15];
  const float* bd2 = (const float*)d_in[16];
  const float* Wa  = (const float*)d_in[17];
  const float* ba  = (const float*)d_in[18];
  const float* Wb  = (const float*)d_in[19];
  const float* bb  = (const float*)d_in[20];
  const float* tau = (const float*)d_in[21];
  float* outf = (float*)d_out;

  // workspace carve-up (floats); heavy reuse of the m1 region after scatter
  float* w    = (float*)d_ws;
  const size_t NR = (size_t)kN;
  float* m1   = w;                 // [N,256] encoder pre-scatter
  float* agg1 = w + NR * 256;      // [N,256] -> h1 (in-place LN+ReLU)
  float* mm   = w;                 // [N,64]  reuse of m1 region
  float* lm   = w + NR * 64;       // [N,64]
  float* muA  = w + NR * 128;      // [N,64]
  float* lvA  = w + NR * 192;      // [N,64]
  float* zbuf = w + NR * 512;      // [N,64]
  float* md   = zbuf + NR * 64;    // [N,64]
  float* hbuf = md + NR * 64;      // [N,64]
  float* dis  = hbuf + NR * 64;    // [N]
  float* acc  = dis + NR;          // [8] scalar accumulators
  float* W1P  = acc + 8;           // packed weights (fragment order)
  float* WmuP = W1P + 65536;       // 256x256 -> 65536
  float* WlvP = WmuP + 16384;      // 256x64  -> 16384
  float* Wd1P = WlvP + 16384;
  float* Wd2P = Wd1P + 4096;       // 64x64   -> 4096
  float* WaP  = Wd2P + 4096;       // 256x64  -> 16384
  float* md2  = mm;                // reuse (mm dead after mu/lv scatter)
  float* h2   = lm;                // reuse

  const int* src = ei;             // edge_index row 0
  const int* dst = ei + kE2;       // edge_index row 1
  const int* ps  = pos;            // pos_edge row 0
  const int* pd  = pos + kE;       // pos_edge row 1

  // --- pack all weight matrices into WMMA fragment order ---
  pack_b_frags<<<(65536 + 255) / 256, 256, 0, stream>>>(W1, W1P, 256, 256);
  pack_b_frags<<<(8192 + 255) / 256, 256, 0, stream>>>(Wmu, WmuP, 256, 64);
  pack_b_frags<<<(8192 + 255) / 256, 256, 0, stream>>>(Wlv, WlvP, 256, 64);
  pack_b_frags<<<(2048 + 255) / 256, 256, 0, stream>>>(Wd1, Wd1P, 64, 64);
  pack_b_frags<<<(2048 + 255) / 256, 256, 0, stream>>>(Wd2, Wd2P, 64, 64);
  pack_b_frags<<<(8192 + 255) / 256, 256, 0, stream>>>(Wa, WaP, 256, 64);

  // --- encoder ---
  hipMemsetAsync(agg1, 0, NR * 256 * sizeof(float), stream);
  {
    int tiles = (kN / 32) * (256 / 16);   // 10000
    gemm_wmma_f32<<<(tiles + 3) / 4, 128, 0, stream>>>(x, W1P, m1, kN, 256, 256);
  }
  scatter_add_256<<<(kE2 * 64 + 255) / 256, 256, 0, stream>>>(m1, agg1, src, dst);
  ln_relu_rows<<<kN / 8, 256, 0, stream>>>(agg1, b1, g1, bt1);
  {
    int tiles = (kN / 32) * (64 / 16);    // 2500
    gemm_wmma_f32<<<(tiles + 3) / 4, 128, 0, stream>>>(agg1, WmuP, mm, kN, 64, 256);
    gemm_wmma_f32<<<(tiles + 3) / 4, 128, 0, stream>>>(agg1, WlvP, lm, kN, 64, 256);
  }
  hipMemsetAsync(muA, 0, NR * 128 * sizeof(float), stream);  // muA+lvA contiguous
  scatter_add_64x2<<<(kE2 * 16 + 255) / 256, 256, 0, stream>>>(mm, lm, muA, lvA, src, dst);
  hipMemsetAsync(acc, 0, 8 * sizeof(float), stream);
  z_and_kl<<<(kN * 64 + 255) / 256, 256, 0, stream>>>(muA, lvA, eps, bmu, blv, zbuf, acc);

  // --- decoder normalization ---
  hipMemsetAsync(dis, 0, NR * sizeof(float), stream);
  deg_count<<<(kE + 255) / 256, 256, 0, stream>>>(pd, dis);
  deg_finish<<<(kN + 255) / 256, 256, 0, stream>>>(dis);

  // --- decoder layer 1 ---
  {
    int tiles = (kN / 32) * (64 / 16);
    gemm_wmma_f32<<<(tiles + 3) / 4, 128, 0, stream>>>(zbuf, Wd1P, md, kN, 64, 64);
  }
  hipMemsetAsync(hbuf, 0, NR * 64 * sizeof(float), stream);
  scatter_norm64<<<(kE * 16 + 255) / 256, 256, 0, stream>>>(md, hbuf, ps, pd, dis);
  self_bias_relu<<<(kN * 64 + 255) / 256, 256, 0, stream>>>(hbuf, md, dis, bd1);

  // --- decoder layer 2 ---
  {
    int tiles = (kN / 32) * (64 / 16);
    gemm_wmma_f32<<<(tiles + 3) / 4, 128, 0, stream>>>(hbuf, Wd2P, md2, kN, 64, 64);
  }
  hipMemsetAsync(h2, 0, NR * 64 * sizeof(float), stream);
  scatter_norm64<<<(kE * 16 + 255) / 256, 256, 0, stream>>>(md2, h2, ps, pd, dis);
  self_bias_relu<<<(kN * 64 + 255) / 256, 256, 0, stream>>>(h2, md2, dis, bd2);

  // --- edge MLP + BCE (dominant GEMM, WMMA) ---
  edge_mlp_wmma<<<kEGRP, 32, 0, stream>>>(h2, pos, neg, WaP, ba, Wb, bb, tau, acc);

  finalize_loss<<<1, 1, 0, stream>>>(acc, outf);
}